// regionAgg_layer_65867618451490
// MI455X (gfx1250) — compile-verified
//
#include <hip/hip_runtime.h>

typedef __attribute__((ext_vector_type(2))) float v2f;
typedef __attribute__((ext_vector_type(8))) float v8f;

#define BATCH 8
#define NPAR  256
#define HWPX  65536
#define TILES (BATCH * (NPAR / 16))   // 128 tiles of 16 parcels

// One workgroup = 8 waves. Each wave runs a V_WMMA_F32_16X16X4_F32 MAC loop:
//   A (16x4 f32)  = mask[n0..n0+15, h..h+3]  (lane l -> row n0+l%16, K-pair 2*(l/16))
//   B (4x16 f32)  = x[h..h+3] broadcast across the 16 columns (same lane/K split)
//   D (16x16 f32) accumulates; every column identical = parcel partial dots.
// Mask is streamed non-temporally (512MB > 192MB L2); x stays cached (2MB, 256x reuse).
__global__ __launch_bounds__(256) void parcel_agg_wmma(
    const float* __restrict__ x, const float* __restrict__ mask,
    float* __restrict__ part, int chunks, int niter) {
  __shared__ float red[8 * 16];

  const int tile  = blockIdx.x;          // 0..127
  const int chunk = blockIdx.y;          // 0..chunks-1
  const int b     = tile >> 4;
  const int n0    = (tile & 15) * 16;
  const int lane  = threadIdx.x & 31;
  const int wave  = threadIdx.x >> 5;    // 0..7
  const int row   = n0 + (lane & 15);
  const int koff  = (lane >> 4) * 2;     // lower half K=0,1 ; upper half K=2,3

  // niter = hlen/32 : uniform scalar trip count (>=32, multiple of 8)
  __builtin_assume(niter >= 8 && (niter & 7) == 0);

  const int hstart = chunk * (niter * 32) + wave * 4;
  const float* pm = mask + ((size_t)(b * NPAR + row)) * HWPX + hstart + koff;
  const float* px = x + (size_t)b * HWPX + hstart + koff;

  v8f acc = {};
#pragma unroll 4
  for (int i = 0; i < niter; ++i) {
    // Non-temporal streaming load of the mask tile row-pair.
    v2f a  = __builtin_nontemporal_load((const v2f*)(pm + (size_t)i * 32));
    v2f bv = *(const v2f*)(px + (size_t)i * 32);   // x pair, cache-resident
    acc = __builtin_amdgcn_wmma_f32_16x16x4_f32(
        /*neg_a=*/false, a, /*neg_b=*/false, bv,
        /*c_mod=*/(short)0, acc, /*reuse_a=*/false, /*reuse_b=*/false);
  }

  // C/D layout: VGPR r, lanes 0-15 -> M=r ; lanes 16-31 -> M=8+r. Columns identical,
  // so lane 0 holds parcels n0+0..7 and lane 16 holds parcels n0+8..15.
  if (lane == 0 || lane == 16) {
    const int half = lane >> 4;
#pragma unroll
    for (int m = 0; m < 8; ++m)
      red[wave * 16 + half * 8 + m] = acc[m];
  }
  __syncthreads();

  // Combine the 8 waves' partials; one partial per (tile, parcel, chunk).
  if (threadIdx.x < 16) {
    float s = 0.f;
#pragma unroll
    for (int w = 0; w < 8; ++w) s += red[w * 16 + threadIdx.x];
    part[((size_t)(tile * 16 + threadIdx.x)) * chunks + chunk] = s;
  }
}

// Deterministic second pass: sum the per-chunk partials (no float atomics).
__global__ void parcel_agg_reduce(const float* __restrict__ part,
                                  float* __restrict__ out, int chunks) {
  const int i = blockIdx.x * blockDim.x + threadIdx.x;
  if (i < BATCH * NPAR) {
    float s = 0.f;
    for (int c = 0; c < chunks; ++c) s += part[(size_t)i * chunks + c];
    out[i] = s;
  }
}

extern "C" void kernel_launch(void* const* d_in, const int* in_sizes, int n_in,
                              void* d_out, int out_size, void* d_ws, size_t ws_size,
                              hipStream_t stream) {
  const float* x    = (const float*)d_in[0];  // [8, 65536] f32
  const float* mask = (const float*)d_in[1];  // [8, 256, 65536] f32
  float* out        = (float*)d_out;          // [8, 256] f32
  (void)in_sizes; (void)n_in; (void)out_size;

  // How many per-chunk partial slices fit in the workspace (each 2048 floats).
  const size_t slice = (size_t)BATCH * NPAR * sizeof(float);
  int cap = (int)(ws_size / slice);
  int chunks = 1;
  while (chunks * 2 <= cap && chunks < 64) chunks *= 2;  // power of two, <= 64

  const bool use_ws = (cap >= 1);
  float* part = use_ws ? (float*)d_ws : out;   // fallback: single-chunk direct write
  if (!use_ws) chunks = 1;
  const int hlen  = HWPX / chunks;             // multiple of 1024
  const int niter = hlen / 32;                 // per-wave WMMA iterations (>=32, mult of 8)

  dim3 grid(TILES, chunks);
  parcel_agg_wmma<<<grid, 256, 0, stream>>>(x, mask, part, chunks, niter);
  if (use_ws) {
    parcel_agg_reduce<<<(BATCH * NPAR + 255) / 256, 256, 0, stream>>>(part, out, chunks);
  }
}